// CaptionModule_24137716203571
// MI455X (gfx1250) — compile-verified
//
#include <hip/hip_runtime.h>
#include <hip/hip_bf16.h>
#include <cstdint>

#define BB   32
#define TT   20
#define BEAM 5
#define VV   32000
#define HH   512
#define NEGV (-1000.0f)

typedef float v4f __attribute__((ext_vector_type(4)));

#if defined(__AMDGCN__)
  #if __has_builtin(__builtin_amdgcn_global_load_async_to_lds_b128)
    #define HAVE_ASYNC_LDS 1
    typedef int v4i_t __attribute__((vector_size(16)));
    typedef __attribute__((address_space(1))) v4i_t gv4i;   // global
    typedef __attribute__((address_space(3))) v4i_t lv4i;   // LDS
  #endif
  #if __has_builtin(__builtin_amdgcn_s_wait_asynccnt)
    #define WAIT_ASYNC(n) __builtin_amdgcn_s_wait_asynccnt(n)
  #else
    #define WAIT_ASYNC(n) asm volatile("s_wait_asynccnt %0" :: "n"(n) : "memory")
  #endif
#endif

// ---------------------------------------------------------------------------
// Phase 1: per-(b,t,beam) top-5 over V=32000 (with unk penalty applied).
// One block (8 wave32s) per row. Each wave streams chunks of 128 floats
// (32 lanes x float4) through a private double-buffered LDS tile using
// CDNA5 async global->LDS loads; each lane maintains a sorted top-5 in
// registers; LDS tree-merge produces the row's top-5 (values + indices).
// ---------------------------------------------------------------------------
__global__ __launch_bounds__(256) void k_top5(const float* __restrict__ lp,
                                              const int* __restrict__ unkp,
                                              float* __restrict__ topv,
                                              int* __restrict__ topi)
{
    const int r    = blockIdx.x;          // 0..3199 == ((b*TT)+t)*BEAM + beam
    const int tid  = threadIdx.x;         // 0..255
    const int wave = tid >> 5;
    const int lane = tid & 31;
    const int unk  = unkp[0];
    const float* row = lp + (size_t)r * VV;

    float bv[5];
    int   bi[5];
#pragma unroll
    for (int k = 0; k < 5; ++k) { bv[k] = -__builtin_huge_valf(); bi[k] = 0; }

    // sorted-descending insertion; strict '>' keeps earlier index on ties
    auto ins5 = [&](float x, int gi) {
        if (x > bv[4]) {
            bv[4] = x; bi[4] = gi;
            if (bv[4] > bv[3]) { float tv=bv[3]; bv[3]=bv[4]; bv[4]=tv; int ti=bi[3]; bi[3]=bi[4]; bi[4]=ti; }
            if (bv[3] > bv[2]) { float tv=bv[2]; bv[2]=bv[3]; bv[3]=tv; int ti=bi[2]; bi[2]=bi[3]; bi[3]=ti; }
            if (bv[2] > bv[1]) { float tv=bv[1]; bv[1]=bv[2]; bv[2]=tv; int ti=bi[1]; bi[1]=bi[2]; bi[2]=ti; }
            if (bv[1] > bv[0]) { float tv=bv[0]; bv[0]=bv[1]; bv[1]=tv; int ti=bi[0]; bi[0]=bi[1]; bi[1]=ti; }
        }
    };

    auto process = [&](v4f x, int gbase) {
#pragma unroll
        for (int j = 0; j < 4; ++j) {
            float xv = x[j];
            int   gi = gbase + j;
            if (gi == unk) xv += NEGV;   // lp.at[:, unk].add(NEG)
            ins5(xv, gi);
        }
    };

    // 32000 floats = 250 chunks of 128; wave w takes chunks c == w (mod 8)
#if defined(HAVE_ASYNC_LDS)
    __shared__ __align__(16) float sbuf[8][2][128];   // per-wave double buffer
    auto issue = [&](int c, int bufi) {
        const float* g = row + c * 128 + lane * 4;
        float*       l = &sbuf[wave][bufi][lane * 4];
        __builtin_amdgcn_global_load_async_to_lds_b128((gv4i*)g, (lv4i*)l, 0, 0);
    };
    int c = wave;
    int bufi = 0;
    if (c < 250) issue(c, 0);
    while (c < 250) {
        int nc = c + 8;
        if (nc < 250) { issue(nc, bufi ^ 1); WAIT_ASYNC(1); }
        else          { WAIT_ASYNC(0); }
        v4f x = *(const v4f*)&sbuf[wave][bufi][lane * 4];
        process(x, c * 128 + lane * 4);
        c = nc; bufi ^= 1;
    }
#else
    for (int c = wave; c < 250; c += 8) {
        const v4f* p = (const v4f*)(row + c * 128 + lane * 4);
        v4f x = __builtin_nontemporal_load(p);   // streaming: read-once data
        process(x, c * 128 + lane * 4);
    }
#endif

    // tree-merge 256 sorted 5-lists -> row top-5
    __shared__ float rv[256][5];
    __shared__ int   ri[256][5];
#pragma unroll
    for (int k = 0; k < 5; ++k) { rv[tid][k] = bv[k]; ri[tid][k] = bi[k]; }
    __syncthreads();
    for (int s = 128; s > 0; s >>= 1) {
        if (tid < s) {
            float ov[5]; int oi[5];
            int a = 0, b2 = 0;
#pragma unroll
            for (int k = 0; k < 5; ++k) {
                float va = rv[tid][a],     vb = rv[tid + s][b2];
                int   ia = ri[tid][a],     ib = ri[tid + s][b2];
                bool ta = (va > vb) || (va == vb && ia <= ib);   // lower index wins ties
                ov[k] = ta ? va : vb;
                oi[k] = ta ? ia : ib;
                if (ta) ++a; else ++b2;
            }
#pragma unroll
            for (int k = 0; k < 5; ++k) { rv[tid][k] = ov[k]; ri[tid][k] = oi[k]; }
        }
        __syncthreads();
    }
    if (tid == 0) {
#pragma unroll
        for (int k = 0; k < 5; ++k) {
            topv[r * 5 + k] = rv[0][k];
            topi[r * 5 + k] = ri[0][k];
        }
    }
}

// ---------------------------------------------------------------------------
// Phase 2: beam recombination. One wave32 per batch; lanes 0..24 hold the
// 5x5 candidate matrix (lane = src_beam*5 + cand). Five shfl_xor argmax
// reductions per step with lower-flat-index tie-break (matches lax.top_k).
// Records backpointers/tokens/logps in LDS; composes the state permutation;
// backtracks to emit seq / seq_lp in the reference's final-beam order.
// ---------------------------------------------------------------------------
__global__ __launch_bounds__(32) void k_beam(const float* __restrict__ topv,
                                             const int* __restrict__ topi,
                                             const int* __restrict__ eosp,
                                             float* __restrict__ out,
                                             int* __restrict__ permout)
{
    const int b    = blockIdx.x;      // batch
    const int lane = threadIdx.x;     // 0..31
    const int eos  = eosp[0];

    __shared__ float s_lpsum[BEAM];
    __shared__ int   s_perm[BEAM];
    __shared__ int   s_bp[TT][BEAM];
    __shared__ int   s_tok[TT][BEAM];
    __shared__ float s_lgp[TT][BEAM];

    if (lane < BEAM) { s_lpsum[lane] = 0.0f; s_perm[lane] = lane; }
    __syncthreads();

    const int bm = lane / BEAM;       // source beam
    const int cc = lane % BEAM;       // candidate within beam

    for (int t = 0; t < TT; ++t) {
        float tv = 0.0f; int ti = 0;
        float cand = -__builtin_huge_valf();
        if (lane < BEAM * BEAM) {
            const int rr = (b * TT + t) * BEAM + bm;
            tv = topv[rr * BEAM + cc];
            ti = topi[rr * BEAM + cc];
            cand = s_lpsum[bm] + tv;
            if (t == 0 && bm > 0) cand = -__builtin_huge_valf();
        }

        float nsum[BEAM]; int nbp[BEAM]; int ntok[BEAM]; float nlgp[BEAM];
#pragma unroll
        for (int j = 0; j < BEAM; ++j) {
            float v = cand; int idx = lane;
#pragma unroll
            for (int off = 16; off > 0; off >>= 1) {   // wave32: 5 levels
                float ovv = __shfl_xor(v, off, 32);
                int   oii = __shfl_xor(idx, off, 32);
                if (ovv > v || (ovv == v && oii < idx)) { v = ovv; idx = oii; }
            }
            const int   wb  = idx / BEAM;
            const float wtv = __shfl(tv, idx, 32);
            const int   wti = __shfl(ti, idx, 32);
            float ns = s_lpsum[wb] + wtv;
            if (wti == eos) ns = NEGV;
            nsum[j] = ns; nbp[j] = wb; ntok[j] = wti; nlgp[j] = wtv;
            if (lane == idx) cand = -__builtin_huge_valf();   // remove winner
        }
        __syncthreads();
        int np = 0; float myns = 0.0f; int mybp = 0, mytok = 0; float mylgp = 0.0f;
        if (lane < BEAM) {
            mybp = nbp[lane]; mytok = ntok[lane]; mylgp = nlgp[lane]; myns = nsum[lane];
            np = s_perm[mybp];                   // compose state permutation
        }
        __syncthreads();
        if (lane < BEAM) {
            s_lpsum[lane] = myns;
            s_perm[lane]  = np;
            s_bp[t][lane]  = mybp;
            s_tok[t][lane] = mytok;
            s_lgp[t][lane] = mylgp;
        }
        __syncthreads();
    }

    // out layout: seq[0,3200) seq_lp[3200,6400) lp_sum[6400,6560) state[6560,...)
    if (lane < BEAM) {
        out[6400 + b * BEAM + lane] = s_lpsum[lane];
        permout[b * BEAM + lane]    = s_perm[lane];
        int ptr = lane;
        for (int t = TT - 1; t >= 0; --t) {
            out[(b * BEAM + lane) * TT + t]        = (float)s_tok[t][ptr];
            out[3200 + (b * BEAM + lane) * TT + t] = s_lgp[t][ptr];
            ptr = s_bp[t][ptr];
        }
    }
}

// ---------------------------------------------------------------------------
// Phase 3: gather state through the composed permutation.
// state: (B, 2, 1, BEAM, H) -> out[b,s,0,j,h] = in[b,s,0,perm[b][j],h]
// ---------------------------------------------------------------------------
__global__ __launch_bounds__(256) void k_gather(const float* __restrict__ st,
                                                const int* __restrict__ perm,
                                                float* __restrict__ outst)
{
    const int i  = blockIdx.x * 256 + threadIdx.x;   // < 163840
    const int h  = i & (HH - 1);
    const int rem = i / HH;            // (b*2+s)*BEAM + j
    const int j  = rem % BEAM;
    const int bs = rem / BEAM;         // b*2+s
    const int b  = bs >> 1;
    const int pj = perm[b * BEAM + j];
    outst[i] = st[(bs * BEAM + pj) * HH + h];
}

extern "C" void kernel_launch(void* const* d_in, const int* in_sizes, int n_in,
                              void* d_out, int out_size, void* d_ws, size_t ws_size,
                              hipStream_t stream)
{
    (void)in_sizes; (void)n_in; (void)out_size; (void)ws_size;
    const float* lp  = (const float*)d_in[0];   // (32,20,5,32000) f32
    const float* st  = (const float*)d_in[1];   // (32,2,1,5,512)  f32
    const int*   eos = (const int*)d_in[2];     // scalar
    const int*   unk = (const int*)d_in[3];     // scalar
    float* out = (float*)d_out;

    const int nrows = BB * TT * BEAM;           // 3200
    float* topv = (float*)d_ws;                                             // 3200*5 f32
    int*   topi = (int*)((char*)d_ws + (size_t)nrows * BEAM * sizeof(float));
    int*   perm = (int*)((char*)d_ws + (size_t)nrows * BEAM * 8);           // 32*5 int

    k_top5  <<<nrows, 256, 0, stream>>>(lp, unk, topv, topi);
    k_beam  <<<BB,     32, 0, stream>>>(topv, topi, eos, out, perm);
    k_gather<<<(BB * 2 * BEAM * HH) / 256, 256, 0, stream>>>(st, perm, out + 6560);
}